// Model_GCN_13151189860858
// MI455X (gfx1250) — compile-verified
//
#include <hip/hip_runtime.h>

#define D_IN 128
#define D_OUT 64
#define SW_STRIDE 72   // padded LDS stride: 2*72 = 144 == 16 (mod 64 banks) -> conflict-free B reads

typedef __attribute__((ext_vector_type(2))) float v2f;
typedef __attribute__((ext_vector_type(8))) float v8f;

// --- 1. init degree with self-loop contribution -----------------------------
__global__ void k_deg_init(float* __restrict__ deg, int N) {
  int i = blockIdx.x * blockDim.x + threadIdx.x;
  if (i < N) deg[i] = 1.0f;
}

// --- 2. scatter-add ones on dst --------------------------------------------
__global__ void k_deg_edges(const int* __restrict__ ei, float* __restrict__ deg, int E) {
  int e = blockIdx.x * blockDim.x + threadIdx.x;
  if (e < E) atomicAdd(deg + ei[E + e], 1.0f);
}

// --- 3. deg -> deg^-1/2 in place (deg >= 1 always, no zero branch needed) ---
__global__ void k_rsqrt(float* __restrict__ deg, int N) {
  int i = blockIdx.x * blockDim.x + threadIdx.x;
  if (i < N) deg[i] = rsqrtf(deg[i]);
}

// --- 4. h = x @ W via V_WMMA_F32_16X16X4_F32, W staged in LDS ---------------
// One wave per 16-row block of h. 4 accumulators cover the 64 output columns.
// A frag (16x4 f32): lane<16 -> {K=k,k+1}, lane>=16 -> {K=k+2,k+3}, row = lane%16.
// B frag (4x16 f32): same K split, col = lane%16, read from padded LDS copy of W.
// C/D frag: VGPR r -> M = r + 8*(lane>=16), N = lane%16.
__global__ void __launch_bounds__(256)
k_gemm_wmma(const float* __restrict__ x, const float* __restrict__ W,
            float* __restrict__ h, int N) {
  __shared__ float sW[D_IN * SW_STRIDE];   // 36 KB of the WGP's 320 KB LDS

  // cooperative staging of W (32 KB) with b128 loads; row base r*72*4 = r*288 B
  // is 16B-aligned so float4 stores into the padded layout are legal.
  {
    const float4* __restrict__ Wv = (const float4*)W;
    for (int i = threadIdx.x; i < (D_IN * D_OUT) / 4; i += 256) {
      int r = i >> 4, c4 = i & 15;
      *(float4*)&sW[r * SW_STRIDE + c4 * 4] = Wv[i];
    }
  }
  __syncthreads();

  const int wave = (blockIdx.x * blockDim.x + threadIdx.x) >> 5;
  const int lane = threadIdx.x & 31;
  const int row0 = wave * 16;
  if (row0 >= N) return;                 // wave-uniform: EXEC stays all-ones
  const int half = lane >> 4;            // 0 or 1
  const int l    = lane & 15;

  v8f acc0 = {}, acc1 = {}, acc2 = {}, acc3 = {};
  const float* __restrict__ xr = x + (size_t)(row0 + l) * D_IN;

#pragma unroll 4
  for (int k = 0; k < D_IN; k += 4) {
    const int ka = k + 2 * half;
    v2f a;
    a.x = xr[ka];
    a.y = xr[ka + 1];

    const float* __restrict__ w0 = sW + ka * SW_STRIDE + l;   // K row ka
    v2f b0, b1, b2, b3;                                       // .y = K row ka+1
    b0.x = w0[0];  b0.y = w0[SW_STRIDE + 0];
    b1.x = w0[16]; b1.y = w0[SW_STRIDE + 16];
    b2.x = w0[32]; b2.y = w0[SW_STRIDE + 32];
    b3.x = w0[48]; b3.y = w0[SW_STRIDE + 48];

    acc0 = __builtin_amdgcn_wmma_f32_16x16x4_f32(false, a, false, b0, (short)0, acc0, false, false);
    acc1 = __builtin_amdgcn_wmma_f32_16x16x4_f32(false, a, false, b1, (short)0, acc1, false, false);
    acc2 = __builtin_amdgcn_wmma_f32_16x16x4_f32(false, a, false, b2, (short)0, acc2, false, false);
    acc3 = __builtin_amdgcn_wmma_f32_16x16x4_f32(false, a, false, b3, (short)0, acc3, false, false);
  }

  float* __restrict__ hb = h + (size_t)row0 * D_OUT;
#pragma unroll
  for (int r = 0; r < 8; ++r) {
    float* __restrict__ dst = hb + (size_t)(r + 8 * half) * D_OUT + l;
    dst[0]  = acc0[r];
    dst[16] = acc1[r];
    dst[32] = acc2[r];
    dst[48] = acc3[r];
  }
}

// --- 5. out = h * dinv^2 (self loop term; also initializes poisoned d_out) --
// b128 in / b128 out; 16 float4 per node row.
__global__ void k_self_loop(const float4* __restrict__ h4, const float* __restrict__ dinv,
                            float4* __restrict__ out4, int total4) {
  int i = blockIdx.x * blockDim.x + threadIdx.x;
  if (i < total4) {
    float di = dinv[i >> 4];             // D_OUT/4 == 16 float4 per row
    float s  = di * di;
    float4 v = h4[i];
    v.x *= s; v.y *= s; v.z *= s; v.w *= s;
    out4[i] = v;
  }
}

// --- 6. one wave per edge: out[dst] += h[src] * norm (64 contiguous floats) -
// One b64 load per lane; 64 scalar f32 atomics per edge (irreducible).
__global__ void __launch_bounds__(256)
k_scatter(const int* __restrict__ ei, const float* __restrict__ h,
          const float* __restrict__ dinv, float* __restrict__ out, int E) {
  int e = (blockIdx.x * blockDim.x + threadIdx.x) >> 5;
  if (e >= E) return;
  int lane = threadIdx.x & 31;
  int s = ei[e];
  int d = ei[E + e];
  float nrm = dinv[s] * dinv[d];
  const float2* __restrict__ hs2 = (const float2*)(h + (size_t)s * D_OUT);
  float* __restrict__ od = out + (size_t)d * D_OUT;
  float2 v = hs2[lane];
  atomicAdd(od + 2 * lane,     v.x * nrm);
  atomicAdd(od + 2 * lane + 1, v.y * nrm);
}

extern "C" void kernel_launch(void* const* d_in, const int* in_sizes, int n_in,
                              void* d_out, int out_size, void* d_ws, size_t ws_size,
                              hipStream_t stream) {
  const float* x  = (const float*)d_in[0];
  const float* W  = (const float*)d_in[1];
  const int*   ei = (const int*)d_in[2];
  float*       out = (float*)d_out;

  const int N = in_sizes[0] / D_IN;   // 100000
  const int E = in_sizes[2] / 2;      // 3200000

  // workspace layout: [deg/dinv : N floats][h : N*64 floats], h 256B-aligned
  float* deg = (float*)d_ws;
  size_t degBytes = (((size_t)N * sizeof(float)) + 255) & ~(size_t)255;
  float* h = (float*)((char*)d_ws + degBytes);

  const int T = 256;
  k_deg_init <<<(N + T - 1) / T, T, 0, stream>>>(deg, N);
  k_deg_edges<<<(E + T - 1) / T, T, 0, stream>>>(ei, deg, E);
  k_rsqrt    <<<(N + T - 1) / T, T, 0, stream>>>(deg, N);

  int waves  = (N + 15) / 16;         // one wave per 16 output rows
  int blocks = (waves + 7) / 8;       // 8 waves per 256-thread block
  k_gemm_wmma<<<blocks, T, 0, stream>>>(x, W, h, N);

  int total4 = (N * D_OUT) / 4;
  k_self_loop<<<(total4 + T - 1) / T, T, 0, stream>>>((const float4*)h, deg,
                                                      (float4*)out, total4);

  int sblocks = (int)(((long long)E * 32 + T - 1) / T);
  k_scatter<<<sblocks, T, 0, stream>>>(ei, h, deg, out, E);
}